// DenseOSTL_60894046323057
// MI455X (gfx1250) — compile-verified
//
#include <hip/hip_runtime.h>

typedef __attribute__((ext_vector_type(2))) float v2f;
typedef __attribute__((ext_vector_type(4))) float v4f;
typedef __attribute__((ext_vector_type(8))) float v8f;

#define BETA   0.95f
#define THRESH 1.0f
#define B_DIM  64
#define D_DIM  1024
#define H_DIM  1024

// -----------------------------------------------------------------------------
// Kernel 1: fused GEMM (h = x@W) + LIF update via V_WMMA_F32_16X16X4_F32.
// One wave32 computes one 16x16 tile of h, then applies:
//   u      = beta*Vmem + h + b
//   spikes = (u - 1 > 0)
//   Vmem'  = u - spikes
//   E_b'   = beta*E_b + 1
// Tiles: (64/16) x (1024/16) = 4 x 64 = 256 waves total.
// -----------------------------------------------------------------------------
__global__ __launch_bounds__(256) void gemm_lif_kernel(
    const float* __restrict__ W,       // [D, H]
    const float* __restrict__ bias,    // [H]
    const float* __restrict__ Vmem,    // [B, H]
    const float* __restrict__ E_b,     // [B, H]
    const float* __restrict__ x,       // [B, D]
    float* __restrict__ Vmem_new,      // [B, H]
    float* __restrict__ E_b_new,       // [B, H]
    float* __restrict__ spikes)        // [B, H]
{
    const int lane  = threadIdx.x & 31;
    const int wave  = blockIdx.x * (blockDim.x >> 5) + (threadIdx.x >> 5);
    const int nt    = wave & (H_DIM / 16 - 1);   // 0..63  (N tile)
    const int mt    = wave >> 6;                 // 0..3   (M tile)
    const int m0    = mt * 16;
    const int n0    = nt * 16;
    const int half  = lane >> 4;                 // 0: lanes 0-15, 1: lanes 16-31
    const int sub   = lane & 15;
    const int mA    = m0 + sub;                  // A-matrix row for this lane
    const int nB    = n0 + sub;                  // B/C/D column for this lane

    v8f acc = {};
    // K loop: 4 elements of K per WMMA. ISA layout (16x4 f32 A):
    //   VGPR0 = K = 2*half + 0, VGPR1 = K = 2*half + 1
    #pragma unroll 4
    for (int k0 = 0; k0 < D_DIM; k0 += 4) {
        const int ka = k0 + 2 * half;
        v2f a, bb;
        a.x  = x[mA * D_DIM + ka];
        a.y  = x[mA * D_DIM + ka + 1];
        bb.x = W[(ka    ) * H_DIM + nB];
        bb.y = W[(ka + 1) * H_DIM + nB];
        acc = __builtin_amdgcn_wmma_f32_16x16x4_f32(
            /*neg_a=*/false, a, /*neg_b=*/false, bb,
            /*c_mod=*/(short)0, acc, /*reuse_a=*/false, /*reuse_b=*/false);
    }

    // Epilogue: C/D VGPR r holds row M = m0 + r + 8*half, col N = n0 + sub.
    const float bv = bias[nB];
    #pragma unroll
    for (int r = 0; r < 8; ++r) {
        const int mm  = m0 + r + 8 * half;
        const int idx = mm * H_DIM + nB;
        const float u = BETA * Vmem[idx] + acc[r] + bv;
        const float s = (u - THRESH) > 0.0f ? 1.0f : 0.0f;
        Vmem_new[idx] = u - s;
        spikes[idx]   = s;
        E_b_new[idx]  = BETA * E_b[idx] + 1.0f;
    }
}

// -----------------------------------------------------------------------------
// Kernel 2: streaming eligibility trace  E_W' = beta*E_W + x[b,d]  (broadcast
// over the contiguous H axis). 512 MB of traffic -> pure HBM bandwidth.
// One block per (b,d) row: 256 threads x 128-bit = 1024 floats.
// Non-temporal 128-bit accesses: stream past L2 (512MB traffic > 192MB L2).
// -----------------------------------------------------------------------------
__global__ __launch_bounds__(256) void ew_stream_kernel(
    const float* __restrict__ E_W,     // [B*D, H]
    const float* __restrict__ x,       // [B*D]  (flat [B,D])
    float* __restrict__ E_W_new)       // [B*D, H]
{
    const int row  = blockIdx.x;               // 0 .. B*D-1
    const float xv = x[row];                   // scalar broadcast per block
    const int base = row * H_DIM + (threadIdx.x << 2);

    const v4f* src = reinterpret_cast<const v4f*>(E_W + base);
    v4f v = __builtin_nontemporal_load(src);
    v.x = BETA * v.x + xv;
    v.y = BETA * v.y + xv;
    v.z = BETA * v.z + xv;
    v.w = BETA * v.w + xv;
    v4f* dst = reinterpret_cast<v4f*>(E_W_new + base);
    __builtin_nontemporal_store(v, dst);
}

// -----------------------------------------------------------------------------
// Launch. Inputs (setup_inputs order): W, b, Vmem, E_W, E_b, x.
// Output (concat, f32): Vmem_new [B*H] | E_W_new [B*D*H] | E_b_new [B*H] | spikes [B*H]
// -----------------------------------------------------------------------------
extern "C" void kernel_launch(void* const* d_in, const int* in_sizes, int n_in,
                              void* d_out, int out_size, void* d_ws, size_t ws_size,
                              hipStream_t stream) {
    (void)in_sizes; (void)n_in; (void)out_size; (void)d_ws; (void)ws_size;

    const float* W    = (const float*)d_in[0];
    const float* b    = (const float*)d_in[1];
    const float* Vmem = (const float*)d_in[2];
    const float* E_W  = (const float*)d_in[3];
    const float* E_b  = (const float*)d_in[4];
    const float* x    = (const float*)d_in[5];

    float* out       = (float*)d_out;
    float* Vmem_new  = out;                                       // B*H
    float* E_W_new   = Vmem_new + (size_t)B_DIM * H_DIM;          // B*D*H
    float* E_b_new   = E_W_new + (size_t)B_DIM * D_DIM * H_DIM;   // B*H
    float* spikes    = E_b_new + (size_t)B_DIM * H_DIM;           // B*H

    // GEMM+LIF: 256 waves = 32 blocks x 256 threads (8 wave32 per block)
    gemm_lif_kernel<<<32, 256, 0, stream>>>(W, b, Vmem, E_b, x,
                                            Vmem_new, E_b_new, spikes);

    // E_W stream: one block per (b,d) row
    ew_stream_kernel<<<B_DIM * D_DIM, 256, 0, stream>>>(E_W, x, E_W_new);
}